// NetVLAD_15418932592835
// MI455X (gfx1250) — compile-verified
//
#include <hip/hip_runtime.h>
#include <hip/hip_bf16.h>

typedef float v2f __attribute__((ext_vector_type(2)));
typedef float v8f __attribute__((ext_vector_type(8)));
typedef unsigned int u32x4 __attribute__((ext_vector_type(4)));
typedef int i32x4 __attribute__((ext_vector_type(4)));
typedef int i32x8 __attribute__((ext_vector_type(8)));

#define B_ 16
#define N_ 16384
#define D_ 256
#define K_ 32
#define NTILES 32
#define NT (N_ / NTILES)           // 512 rows per workgroup
#define WAVES 8
#define ROWS_PER_WAVE (NT / WAVES) // 64
#define MT (ROWS_PER_WAVE / 16)    // 4 wmma M-tiles per wave
#define TILE_ELEMS (16 * D_)       // 4096 floats / 16 KB per x tile
#define EPS 1e-12f

#if defined(__HIP_DEVICE_COMPILE__) && __has_builtin(__builtin_amdgcn_tensor_load_to_lds)
#define USE_TDM 1
#else
#define USE_TDM 0
#endif

// ---------------------------------------------------------------------------
// Issue one TDM load of a 16-row x 256-col fp32 tile (row stride 256) from
// global memory into LDS at byte offset ldsOff. D# layout per CDNA5 ISA §8.
// ---------------------------------------------------------------------------
#if USE_TDM
__device__ __forceinline__ void tdm_load_tile(unsigned int ldsOff,
                                              unsigned long long gaddr) {
  u32x4 g0;
  g0[0] = 1u;                                  // count=1 user descriptor
  g0[1] = ldsOff;                              // lds_addr (bytes)
  g0[2] = (unsigned int)gaddr;                 // global_addr[31:0]  (bits 95:64)
  g0[3] = (unsigned int)((gaddr >> 32) & 0x01FFFFFFu) | (2u << 30); // addr hi | type=2
  i32x8 g1;
  g1[0] = (int)(2u << 16);                     // workgroup_mask=0, data_size=4B
  g1[1] = (int)((D_ & 0xFFFFu) << 16);         // tensor_dim0 lo16 (=256)
  g1[2] = (int)((D_ >> 16) | (16u << 16));     // tensor_dim0 hi | tensor_dim1 lo16 (=16)
  g1[3] = (int)((unsigned)D_ << 16);           // tensor_dim1 hi | tile_dim0 (=256)
  g1[4] = 16;                                  // tile_dim1=16, tile_dim2=0
  g1[5] = D_;                                  // tensor_dim0_stride lo32 (=256)
  g1[6] = 0;                                   // stride hi | tensor_dim1_stride lo
  g1[7] = 0;
  i32x4 z4 = (i32x4){0, 0, 0, 0};
#if defined(__clang_major__) && (__clang_major__ >= 23)
  i32x8 z8 = (i32x8){0, 0, 0, 0, 0, 0, 0, 0};
  __builtin_amdgcn_tensor_load_to_lds(g0, g1, z4, z4, z8, 0);
#else
  __builtin_amdgcn_tensor_load_to_lds(g0, g1, z4, z4, 0);
#endif
}
#endif

// ---------------------------------------------------------------------------
// Stage 1: fused logits -> softmax -> VLAD aggregation (per N-tile partials)
// ---------------------------------------------------------------------------
__global__ __launch_bounds__(256)
void netvlad_stage1(const float* __restrict__ x, const float* __restrict__ W,
                    const float* __restrict__ bias,
                    float* __restrict__ pV, float* __restrict__ pA) {
  // sBuf: W (32x256) during compute, reused as K x D reduction buffer after.
  // sX:  per-wave DOUBLE-buffered 16x256 fp32 x-tiles filled by the TDM.
  // Total static LDS ~= 32KB + 256KB + 256B (fits 320KB WGP LDS).
  __shared__ float sBuf[K_ * D_];
  __shared__ float sX[WAVES][2][TILE_ELEMS];
  __shared__ float sAsum[K_];
  __shared__ float sBias[K_];

  const int tid  = threadIdx.x;
  const int wave = tid >> 5;
  const int lane = tid & 31;
  const int lhalf = lane & 15;
  const bool hi  = lane >= 16;
  const int koff = hi ? 2 : 0;

  const int blk  = blockIdx.x;
  const int b    = blk / NTILES;
  const int tile = blk % NTILES;

  for (int i = tid; i < K_ * D_; i += 256) sBuf[i] = W[i];
  if (tid < K_) { sBias[tid] = bias[tid]; sAsum[tid] = 0.0f; }
  __syncthreads();

  // this wave's 64-row slice of x
  const float* xw =
      x + ((size_t)b * N_ + (size_t)tile * NT + (size_t)wave * ROWS_PER_WAVE) * D_;

#if USE_TDM
  const unsigned int ldsBase = (unsigned int)__builtin_amdgcn_readfirstlane(
      (unsigned int)(uintptr_t)&sX[wave][0][0]);
  const unsigned long long gBase =
      (unsigned long long)__builtin_amdgcn_readfirstlane((unsigned long long)(uintptr_t)xw);
  // prologue: fill both buffers
  tdm_load_tile(ldsBase, gBase);
  if (MT > 1)
    tdm_load_tile(ldsBase + TILE_ELEMS * 4, gBase + (unsigned long long)TILE_ELEMS * 4);
#endif

  // Per-wave K x D fp32 accumulator: 2 k-tiles x 16 d-tiles of 16x16.
  v8f acc[2][16];
#pragma unroll
  for (int kt = 0; kt < 2; ++kt)
#pragma unroll
    for (int dt = 0; dt < 16; ++dt) acc[kt][dt] = (v8f){0,0,0,0,0,0,0,0};

  float asum0 = 0.0f, asum1 = 0.0f;

  for (int mt = 0; mt < MT; ++mt) {
    float* sXt = &sX[wave][mt & 1][0];
#if USE_TDM
    // oldest outstanding DMA (tile mt) must be resident; tile mt+1 may still
    // be streaming into the other buffer.
    if (mt < MT - 1) {
      asm volatile("s_wait_tensorcnt 0x1" ::: "memory");
    } else {
      asm volatile("s_wait_tensorcnt 0x0" ::: "memory");
    }
#else
    {
      const float4* src = (const float4*)(xw + (size_t)mt * TILE_ELEMS);
      float4* dst = (float4*)sXt;
      for (int i = lane; i < TILE_ELEMS / 4; i += 32) dst[i] = src[i];
    }
#endif
    const float* xrow  = sXt + (size_t)lhalf * D_;
    const float* wrow0 = &sBuf[(size_t)lhalf * D_];
    const float* wrow1 = &sBuf[(size_t)(lhalf + 16) * D_];

    // ---- logits: 16 rows x 32 clusters, K-dim = D = 256 in steps of 4 ----
    v8f c0 = (v8f){0,0,0,0,0,0,0,0};
    v8f c1 = (v8f){0,0,0,0,0,0,0,0};
    for (int k = 0; k < D_; k += 4) {
      v2f aF = *(const v2f*)(xrow  + k + koff);   // A: x rows (16x4 layout)
      v2f b0 = *(const v2f*)(wrow0 + k + koff);   // B: W^T cols 0..15
      v2f b1 = *(const v2f*)(wrow1 + k + koff);   // B: W^T cols 16..31
      c0 = __builtin_amdgcn_wmma_f32_16x16x4_f32(false, aF, false, b0,
                                                 (short)0, c0, false, false);
      c1 = __builtin_amdgcn_wmma_f32_16x16x4_f32(false, aF, false, b1,
                                                 (short)0, c1, false, false);
    }

    // ---- bias + softmax over K=32 (spread across 16-lane halves) ----
    float c0a[8], c1a[8];
#pragma unroll
    for (int j = 0; j < 8; ++j) {
      c0a[j] = c0[j] + sBias[lhalf];
      c1a[j] = c1[j] + sBias[16 + lhalf];
    }
#pragma unroll
    for (int j = 0; j < 8; ++j) {
      float m = fmaxf(c0a[j], c1a[j]);
      m = fmaxf(m, __shfl_xor(m, 1, 32));
      m = fmaxf(m, __shfl_xor(m, 2, 32));
      m = fmaxf(m, __shfl_xor(m, 4, 32));
      m = fmaxf(m, __shfl_xor(m, 8, 32));
      float e0 = __expf(c0a[j] - m);
      float e1 = __expf(c1a[j] - m);
      float s = e0 + e1;
      s += __shfl_xor(s, 1, 32);
      s += __shfl_xor(s, 2, 32);
      s += __shfl_xor(s, 4, 32);
      s += __shfl_xor(s, 8, 32);
      float inv = 1.0f / s;
      c0a[j] = e0 * inv;
      c1a[j] = e1 * inv;
      asum0 += c0a[j];
      asum1 += c1a[j];
    }

    // ---- aggregation: acc[k][d] += a^T (16k x 4n-chunks) @ x (n x d) ----
#pragma unroll
    for (int q = 0; q < 4; ++q) {
      const int base = (q & 1) * 4;   // source register base for rows
      const int sh   = q >> 1;        // storage half of the needed rows
      float t0a = __shfl_xor(c0a[base + 0], 16, 32);
      float t2a = __shfl_xor(c0a[base + 2], 16, 32);
      float u1a = __shfl_xor(c0a[base + 1], 16, 32);
      float u3a = __shfl_xor(c0a[base + 3], 16, 32);
      float t0b = __shfl_xor(c1a[base + 0], 16, 32);
      float t2b = __shfl_xor(c1a[base + 2], 16, 32);
      float u1b = __shfl_xor(c1a[base + 1], 16, 32);
      float u3b = __shfl_xor(c1a[base + 3], 16, 32);
      v2f a0F, a1F;
      if (sh == 0) {
        a0F.x = hi ? t2a : c0a[base + 0];
        a0F.y = hi ? u3a : c0a[base + 1];
        a1F.x = hi ? t2b : c1a[base + 0];
        a1F.y = hi ? u3b : c1a[base + 1];
      } else {
        a0F.x = hi ? c0a[base + 2] : t0a;
        a0F.y = hi ? c0a[base + 3] : u1a;
        a1F.x = hi ? c1a[base + 2] : t0b;
        a1F.y = hi ? c1a[base + 3] : u1b;
      }
      // B fragments from the LDS-resident x tile: rows 4q+{koff,koff+1}
      const float* xr0 = sXt + (size_t)(q * 4 + koff) * D_;
      const float* xr1 = xr0 + D_;
#pragma unroll
      for (int dt = 0; dt < 16; ++dt) {
        v2f bF;
        bF.x = xr0[dt * 16 + lhalf];
        bF.y = xr1[dt * 16 + lhalf];
        acc[0][dt] = __builtin_amdgcn_wmma_f32_16x16x4_f32(
            false, a0F, false, bF, (short)0, acc[0][dt], false, false);
        acc[1][dt] = __builtin_amdgcn_wmma_f32_16x16x4_f32(
            false, a1F, false, bF, (short)0, acc[1][dt], false, false);
      }
    }

#if USE_TDM
    // tail-issue tile mt+2 into the buffer we just finished consuming
    if (mt + 2 < MT) {
      asm volatile("s_wait_dscnt 0x0" ::: "memory");  // LDS reads drained
      tdm_load_tile(ldsBase + (unsigned int)(mt & 1) * TILE_ELEMS * 4,
                    gBase + (unsigned long long)(mt + 2) * TILE_ELEMS * 4);
    }
#endif
  }

  // combine the two lane-halves of a_sum (same k = lane&15 in both halves)
  asum0 += __shfl_xor(asum0, 16, 32);
  asum1 += __shfl_xor(asum1, 16, 32);

  // ---- workgroup reduction: reuse sBuf as the K x D accumulator ----
  __syncthreads();  // everyone done reading W
  for (int i = tid; i < K_ * D_; i += 256) sBuf[i] = 0.0f;
  __syncthreads();
  for (int w = 0; w < WAVES; ++w) {
    if (wave == w) {
#pragma unroll
      for (int kt = 0; kt < 2; ++kt)
#pragma unroll
        for (int dt = 0; dt < 16; ++dt)
#pragma unroll
          for (int j = 0; j < 8; ++j) {
            int k = kt * 16 + j + (hi ? 8 : 0);
            int d = dt * 16 + lhalf;
            sBuf[k * D_ + d] += acc[kt][dt][j];
          }
      if (!hi) {
        sAsum[lhalf]      += asum0;
        sAsum[lhalf + 16] += asum1;
      }
    }
    __syncthreads();
  }

  float* pVo = pV + (size_t)blk * K_ * D_;
  for (int i = tid; i < K_ * D_; i += 256) pVo[i] = sBuf[i];
  if (tid < K_) pA[blk * K_ + tid] = sAsum[tid];
}

// ---------------------------------------------------------------------------
// Stage 2: reduce tile partials, subtract a_sum*k_mean, intra + global norm
// ---------------------------------------------------------------------------
__global__ __launch_bounds__(1024)
void netvlad_stage2(const float* __restrict__ pV, const float* __restrict__ pA,
                    const float* __restrict__ kmean, float* __restrict__ out) {
  __shared__ float sAsum[K_];
  __shared__ float sNorm[K_];
  __shared__ float sScale[K_];
  __shared__ float sG;

  const int b = blockIdx.x;
  const int tid = threadIdx.x;

  if (tid < K_) {
    float s = 0.0f;
    for (int t = 0; t < NTILES; ++t) s += pA[(b * NTILES + t) * K_ + tid];
    sAsum[tid] = s;
    sNorm[tid] = 0.0f;
  }
  __syncthreads();

  float v[8];
#pragma unroll
  for (int i = 0; i < 8; ++i) {
    int idx = tid + i * 1024;                 // idx = k*256 + d
    float s = 0.0f;
    for (int t = 0; t < NTILES; ++t)
      s += pV[(size_t)(b * NTILES + t) * (K_ * D_) + idx];
    int k = idx >> 8;
    s -= sAsum[k] * kmean[idx];
    v[i] = s;
    atomicAdd(&sNorm[k], s * s);
  }
  __syncthreads();

  if (tid < K_) {
    float n2 = sNorm[tid];
    float n  = sqrtf(n2);
    float inv = 1.0f / fmaxf(n, EPS);
    sScale[tid] = inv;
    sNorm[tid]  = n2 * inv * inv;   // contribution to global norm^2
  }
  __syncthreads();
  if (tid == 0) {
    float g = 0.0f;
    for (int k = 0; k < K_; ++k) g += sNorm[k];
    sG = 1.0f / fmaxf(sqrtf(g), EPS);
  }
  __syncthreads();

  float gi = sG;
#pragma unroll
  for (int i = 0; i < 8; ++i) {
    int idx = tid + i * 1024;
    int k = idx >> 8;
    out[(size_t)b * (K_ * D_) + idx] = v[i] * sScale[k] * gi;
  }
}

// ---------------------------------------------------------------------------
extern "C" void kernel_launch(void* const* d_in, const int* in_sizes, int n_in,
                              void* d_out, int out_size, void* d_ws, size_t ws_size,
                              hipStream_t stream) {
  (void)in_sizes; (void)n_in; (void)out_size; (void)ws_size;
  const float* x     = (const float*)d_in[0];
  const float* W     = (const float*)d_in[1];
  const float* bias  = (const float*)d_in[2];
  const float* kmean = (const float*)d_in[3];

  float* pV = (float*)d_ws;                                   // B*NTILES*K*D
  float* pA = pV + (size_t)B_ * NTILES * K_ * D_;             // B*NTILES*K

  netvlad_stage1<<<B_ * NTILES, 256, 0, stream>>>(x, W, bias, pV, pA);
  netvlad_stage2<<<B_, 1024, 0, stream>>>(pV, pA, kmean, (float*)d_out);
}